// GCNNet_62955630625290
// MI455X (gfx1250) — compile-verified
//
#include <hip/hip_runtime.h>
#include <hip/hip_bf16.h>

typedef float v2f __attribute__((ext_vector_type(2)));
typedef float v8f __attribute__((ext_vector_type(8)));

#define HID 16

// ---------------------------------------------------------------------------
// degree / norm precompute
// ---------------------------------------------------------------------------
__global__ void k_init_deg(float* __restrict__ deg, int n) {
    int i = blockIdx.x * blockDim.x + threadIdx.x;
    if (i < n) deg[i] = 1.0f;   // self-loop weight
}

__global__ void k_deg_edges(const long long* __restrict__ col,
                            const float* __restrict__ ew,
                            float* __restrict__ deg, long long nE) {
    long long e = (long long)blockIdx.x * blockDim.x + threadIdx.x;
    if (e < nE) atomicAdd(&deg[(int)col[e]], ew[e]);
}

__global__ void k_dinv(const float* __restrict__ deg,
                       float* __restrict__ dinv, int n) {
    int i = blockIdx.x * blockDim.x + threadIdx.x;
    if (i < n) {
        float d = deg[i];
        dinv[i] = (d > 0.0f) ? rsqrtf(d) : 0.0f;
    }
}

__global__ void k_norm(const long long* __restrict__ row,
                       const long long* __restrict__ col,
                       const float* __restrict__ ew,
                       const float* __restrict__ dinv,
                       float* __restrict__ norm, long long nE) {
    long long e = (long long)blockIdx.x * blockDim.x + threadIdx.x;
    if (e < nE) norm[e] = dinv[(int)row[e]] * ew[e] * dinv[(int)col[e]];
}

// ---------------------------------------------------------------------------
// H = act(X) @ W   via V_WMMA_F32_16X16X4_F32  (X: [N,K], W: [K,16], H: [N,16])
// One wave computes one 16-row tile; W staged in LDS in WMMA-B layout:
//   step k, lane l holds B elements (K=4k+2*(l>>4)+{0,1}, N=l&15)
// A layout per ISA: lane 0-15 -> row m, K=4k+{0,1}; lane 16-31 -> K=4k+{2,3}
// ---------------------------------------------------------------------------
template <int K, bool RELU>
__global__ void k_gemm_wmma(const float* __restrict__ X,
                            const float* __restrict__ W,
                            float* __restrict__ H,
                            int n, int nTiles) {
    __shared__ float Bst[(K / 4) * 64];

    const int tid = threadIdx.x;
    for (int idx = tid; idx < K * 16; idx += blockDim.x) {
        int k  = idx >> 6;
        int r  = idx & 63;
        int l  = r >> 1;
        int j  = r & 1;
        int KK = 4 * k + 2 * (l >> 4) + j;
        int NN = l & 15;
        Bst[idx] = W[KK * 16 + NN];
    }
    __syncthreads();

    const int lane = tid & 31;
    const int wave = tid >> 5;
    const int tile = blockIdx.x * (blockDim.x >> 5) + wave;
    if (tile >= nTiles) return;                 // whole wave exits: EXEC stays full for WMMA

    const int half = lane >> 4;                 // 0: K+0/1, 1: K+2/3
    const int m    = tile * 16 + (lane & 15);
    const int mc   = (m < n) ? m : (n - 1);     // clamp OOB loads (stores guarded)
    const float* xrow = X + (long long)mc * K + 2 * half;

    v8f c = {};
#pragma unroll 4
    for (int k = 0; k < K / 4; ++k) {
        v2f a = *(const v2f*)(xrow + 4 * k);
        if (RELU) { a.x = fmaxf(a.x, 0.0f); a.y = fmaxf(a.y, 0.0f); }
        v2f b = *(const v2f*)(&Bst[k * 64 + lane * 2]);
        // 8 args: (neg_a, A, neg_b, B, c_mod, C, reuse_a, reuse_b)
        c = __builtin_amdgcn_wmma_f32_16x16x4_f32(false, a, false, b,
                                                  (short)0, c, false, false);
    }

    // C/D layout: VGPR r, lanes 0-15 -> M=r, lanes 16-31 -> M=r+8
    // Base pointer per lane; full tiles take the unguarded path so the
    // compiler emits 8 immediate-offset global_store_b32 (no exec juggling).
    const int nn = lane & 15;
    float* hp = H + ((long long)tile * 16 + 8 * half) * HID + nn;
    if (tile * 16 + 15 < n) {
#pragma unroll
        for (int r = 0; r < 8; ++r) hp[r * HID] = c[r];
    } else {
#pragma unroll
        for (int r = 0; r < 8; ++r) {
            int mm = tile * 16 + r + 8 * half;
            if (mm < n) hp[r * HID] = c[r];
        }
    }
}

// ---------------------------------------------------------------------------
// Out[i] = bias + self-loop term;  then edge scatter Out[col] += H[row]*norm
// ---------------------------------------------------------------------------
__global__ void k_init_out(const float* __restrict__ H,
                           const float* __restrict__ dinv,
                           const float* __restrict__ bias,
                           float* __restrict__ Out, int n) {
    int idx = blockIdx.x * blockDim.x + threadIdx.x;
    if (idx < n * HID) {
        int i = idx >> 4;
        int f = idx & 15;
        float di = dinv[i];
        Out[idx] = bias[f] + H[idx] * di * di;   // self-loop: norm = dinv^2 * 1
    }
}

__global__ void k_prop_edges(const long long* __restrict__ row,
                             const long long* __restrict__ col,
                             const float* __restrict__ norm,
                             const float* __restrict__ H,
                             float* __restrict__ Out, long long nE) {
    long long t = (long long)blockIdx.x * blockDim.x + threadIdx.x;
    long long e = t >> 4;                        // 16 feature lanes per edge
    if (e >= nE) return;
    int lane = threadIdx.x & 31;
    int f    = lane & 15;
    int src  = lane & 16;                        // half-wave leader: lane 0 / 16
    int r = 0, c = 0; float nm = 0.0f;
    if (f == 0) { r = (int)row[e]; c = (int)col[e]; nm = norm[e]; }
    r  = __shfl(r, src);
    c  = __shfl(c, src);
    nm = __shfl(nm, src);
    atomicAdd(&Out[(long long)c * HID + f], H[(long long)r * HID + f] * nm);
}

// ---------------------------------------------------------------------------
// in-place row-wise log_softmax over 16 values
// ---------------------------------------------------------------------------
__global__ void k_log_softmax(float* __restrict__ Out, int n) {
    int i = blockIdx.x * blockDim.x + threadIdx.x;
    if (i >= n) return;
    float* p = Out + (long long)i * HID;
    float v[HID];
    float m = -3.402823466e+38f;
#pragma unroll
    for (int j = 0; j < HID; ++j) { v[j] = p[j]; m = fmaxf(m, v[j]); }
    float s = 0.0f;
#pragma unroll
    for (int j = 0; j < HID; ++j) s += __expf(v[j] - m);
    float l = m + __logf(s);
#pragma unroll
    for (int j = 0; j < HID; ++j) p[j] = v[j] - l;
}

// ---------------------------------------------------------------------------
static inline size_t alignUp(size_t x) { return (x + 255) & ~(size_t)255; }

extern "C" void kernel_launch(void* const* d_in, const int* in_sizes, int n_in,
                              void* d_out, int out_size, void* d_ws, size_t ws_size,
                              hipStream_t stream) {
    const int       NFEAT = 512;
    const int       N     = in_sizes[0] / NFEAT;        // 100000
    const long long E     = (long long)in_sizes[2];     // 3200000

    const float*     x   = (const float*)d_in[0];
    const long long* ei  = (const long long*)d_in[1];   // int64 [2, E]
    const float*     ew  = (const float*)d_in[2];
    const float*     W1  = (const float*)d_in[3];
    const float*     b1  = (const float*)d_in[4];
    const float*     W2  = (const float*)d_in[5];
    const float*     b2  = (const float*)d_in[6];
    float*           out = (float*)d_out;               // [N,16], also layer-2 accumulator

    const long long* row = ei;
    const long long* col = ei + E;

    // workspace carve-up (~26.4 MB)
    char*  ws  = (char*)d_ws;
    size_t off = 0;
    float* deg  = (float*)(ws + off); off = alignUp(off + (size_t)N * 4);
    float* dinv = (float*)(ws + off); off = alignUp(off + (size_t)N * 4);
    float* nrm  = (float*)(ws + off); off = alignUp(off + (size_t)E * 4);
    float* h0   = (float*)(ws + off); off = alignUp(off + (size_t)N * HID * 4);
    float* out1 = (float*)(ws + off); off = alignUp(off + (size_t)N * HID * 4);
    float* h2   = h0;   // h0 is dead after prop1; reuse for layer-2 features

    const int nTiles = (N + 15) / 16;
    const int gemmBlk   = 256;                  // 8 waves: amortize W staging
    const int gemmGrid  = (nTiles + (gemmBlk >> 5) - 1) / (gemmBlk >> 5);

    // 1) degrees (self-loop = 1) and normalization
    k_init_deg <<<(N + 255) / 256, 256, 0, stream>>>(deg, N);
    k_deg_edges<<<(int)((E + 255) / 256), 256, 0, stream>>>(col, ew, deg, E);
    k_dinv     <<<(N + 255) / 256, 256, 0, stream>>>(deg, dinv, N);
    k_norm     <<<(int)((E + 255) / 256), 256, 0, stream>>>(row, col, ew, dinv, nrm, E);

    // 2) layer 1: h0 = X @ W1 (WMMA f32), aggregate into out1
    k_gemm_wmma<512, false><<<gemmGrid, gemmBlk, 0, stream>>>(x, W1, h0, N, nTiles);
    k_init_out <<<(N * HID + 255) / 256, 256, 0, stream>>>(h0, dinv, b1, out1, N);
    k_prop_edges<<<(int)((E * HID + 255) / 256), 256, 0, stream>>>(row, col, nrm, h0, out1, E);

    // 3) layer 2: h2 = relu(out1) @ W2 (relu fused into A-load), aggregate into d_out
    k_gemm_wmma<16, true><<<gemmGrid, gemmBlk, 0, stream>>>(out1, W2, h2, N, nTiles);
    k_init_out <<<(N * HID + 255) / 256, 256, 0, stream>>>(h2, dinv, b2, out, N);
    k_prop_edges<<<(int)((E * HID + 255) / 256), 256, 0, stream>>>(row, col, nrm, h2, out, E);

    // 4) log_softmax in place
    k_log_softmax<<<(N + 255) / 256, 256, 0, stream>>>(out, N);
}